// GraphConvolution_72189810311698
// MI455X (gfx1250) — compile-verified
//
#include <hip/hip_runtime.h>

// Problem sizes (fixed by the reference)
#define BB 16
#define LL 128
#define DD 256
#define OO 256
#define DQ 64   // DD / 4 (float4 units)

typedef float v2f __attribute__((ext_vector_type(2)));
typedef float v8f __attribute__((ext_vector_type(8)));

// ---------------------------------------------------------------------------
// Kernel 1: U[b,i,d] = sum_j adj[b,i,j] * (text[b,j,d] + dep[b,j,i,d])
// Memory-bound: streams dep_embed (268 MB) exactly once, coalesced along d.
// Block = 256 threads handles one b and 4 consecutive i rows; each 64-thread
// slice owns one i and covers D with float4 loads.
// ---------------------------------------------------------------------------
__global__ __launch_bounds__(256) void gcn_reduce_j_kernel(
    const float* __restrict__ text,   // [BB, LL, DD]
    const float* __restrict__ adj,    // [BB, LL, LL]
    const float* __restrict__ dep,    // [BB, LL, LL, DD]
    float* __restrict__ U)            // [BB, LL, DD]
{
    __shared__ float s_adj[4 * LL];

    const int b     = blockIdx.x >> 5;          // / (LL/4)
    const int itile = (blockIdx.x & 31) << 2;   // 4 i-rows per block
    const int tid   = threadIdx.x;

    // Stage the 4 adj rows (512 floats) into LDS
    for (int idx = tid; idx < 4 * LL; idx += 256) {
        const int r = idx >> 7;       // which i row (0..3)
        const int j = idx & (LL - 1);
        s_adj[idx] = adj[((size_t)(b * LL + itile + r)) * LL + j];
    }
    __syncthreads();

    const int ii = tid >> 6;          // 0..3 -> i row within tile
    const int dq = tid & 63;          // 0..63 -> float4 index along D
    const int i  = itile + ii;

    const float4* text4 = (const float4*)text + (size_t)b * LL * DQ;
    const float4* dep4  = (const float4*)dep  + ((size_t)(b * LL) * LL + i) * DQ;
    const float*  sa    = &s_adj[ii * LL];

    float4 acc = make_float4(0.f, 0.f, 0.f, 0.f);

    #pragma unroll 4
    for (int j = 0; j < LL; ++j) {
        // keep the HBM stream ahead of the FMAs (emits global_prefetch_b8;
        // speculative prefetch past the end)
        __builtin_prefetch(&dep4[(size_t)(j + 16) * (LL * DQ) + dq], 0, 1);

        const float  a = sa[j];
        const float4 t = text4[j * DQ + dq];
        const float4 d = dep4[(size_t)j * (LL * DQ) + dq];
        acc.x += a * (t.x + d.x);
        acc.y += a * (t.y + d.y);
        acc.z += a * (t.z + d.z);
        acc.w += a * (t.w + d.w);
    }

    ((float4*)U)[((size_t)b * LL + i) * DQ + dq] = acc;
}

// ---------------------------------------------------------------------------
// Kernel 2: out[m,o] = relu( U[m,:] @ W[:,o] + bias[o] ),  m = b*LL + i
// M = BB*LL = 2048, N = OO = 256, K = DD = 256.
// Uses V_WMMA_F32_16X16X4_F32 (full f32 precision).
// Block = 256 threads = 8 waves; each wave owns a 16(M) x 64(N) tile
// (4 accumulators, A fragment reused 4x). Grid = (2048/128, 256/64).
//
// VGPR layouts per CDNA5 ISA 7.12.2:
//   A 16x4 : lane l -> M = l%16 ; VGPR {0,1} = K {0,1} (lanes 0-15)
//                                  or K {2,3} (lanes 16-31)
//   B 4x16 : lane l -> N = l%16 ; VGPR {0,1} = K {0,1} / {2,3} by lane half
//   C 16x16: VGPR r -> M = r + 8*(l/16), N = l%16
// ---------------------------------------------------------------------------
__global__ __launch_bounds__(256) void gcn_gemm_bias_relu_kernel(
    const float* __restrict__ U,      // [2048, DD]
    const float* __restrict__ W,      // [DD, OO]
    const float* __restrict__ bias,   // [OO]
    float* __restrict__ out)          // [2048, OO]
{
    const int lane  = threadIdx.x & 31;
    const int wave  = threadIdx.x >> 5;
    const int row0  = blockIdx.x * 128 + wave * 16;   // M tile base for wave
    const int col0  = blockIdx.y * 64;                // N tile base for wave
    const int mn    = lane & 15;                      // M (for A) / N (for B,C)
    const int khalf = lane >> 4;                      // 0: K{0,1}, 1: K{2,3}

    v8f acc[4] = {};

    const float* arow = U + (size_t)(row0 + mn) * DD + 2 * khalf;

    for (int k = 0; k < DD; k += 4) {
        // A fragment: two consecutive K elements of this lane's M row
        const v2f a = *(const v2f*)(arow + k);

        #pragma unroll
        for (int nt = 0; nt < 4; ++nt) {
            const int col = col0 + nt * 16 + mn;
            v2f bf;
            bf.x = W[(size_t)(k + 2 * khalf)     * OO + col];
            bf.y = W[(size_t)(k + 2 * khalf + 1) * OO + col];
            acc[nt] = __builtin_amdgcn_wmma_f32_16x16x4_f32(
                /*neg_a=*/false, a, /*neg_b=*/false, bf,
                /*c_mod=*/(short)0, acc[nt],
                /*reuse_a=*/false, /*reuse_b=*/false);
        }
    }

    // Epilogue: bias + relu, scatter per C layout
    #pragma unroll
    for (int nt = 0; nt < 4; ++nt) {
        const int   col = col0 + nt * 16 + mn;
        const float bv  = bias[col];
        const int   rb  = row0 + 8 * khalf;
        #pragma unroll
        for (int r = 0; r < 8; ++r) {
            float v = acc[nt][r] + bv;
            out[(size_t)(rb + r) * OO + col] = v > 0.f ? v : 0.f;
        }
    }
}

// ---------------------------------------------------------------------------
extern "C" void kernel_launch(void* const* d_in, const int* in_sizes, int n_in,
                              void* d_out, int out_size, void* d_ws, size_t ws_size,
                              hipStream_t stream) {
    const float* text = (const float*)d_in[0];   // [16,128,256]
    const float* adj  = (const float*)d_in[1];   // [16,128,128]
    const float* dep  = (const float*)d_in[2];   // [16,128,128,256]
    const float* W    = (const float*)d_in[3];   // [256,256]
    const float* bias = (const float*)d_in[4];   // [256]
    float*       out  = (float*)d_out;           // [16,128,256]
    float*       U    = (float*)d_ws;            // [16,128,256] scratch (2 MB)

    (void)in_sizes; (void)n_in; (void)out_size; (void)ws_size;

    // Kernel 1: 512 blocks x 256 threads, streams the 268 MB dep_embed tensor
    gcn_reduce_j_kernel<<<BB * (LL / 4), 256, 0, stream>>>(text, adj, dep, U);

    // Kernel 2: WMMA GEMM 2048x256x256 + bias + relu
    dim3 grid2(2048 / 128, OO / 64);
    gcn_gemm_bias_relu_kernel<<<grid2, 256, 0, stream>>>(U, W, bias, out);
}